// cAttention_57784490001265
// MI455X (gfx1250) — compile-verified
//
#include <hip/hip_runtime.h>

// MI455X / gfx1250 fused channel-attention kernel.
// One workgroup (256 threads = 8 wave32) per position p = b*L + l (4096 total).
//  Phase 1: QKV = X_tile(16x256) @ Wqkv^T(256x768) via V_WMMA_F32_16X16X4_F32,
//           q,k,v tiles land in LDS (q gets +bh folded in).
//  Phase 2: additive scores s[i][j] = sum_d Vw[d]*tanh(q[i,d]+k[j,d]), one
//           (i,j) pair per thread (16x16 = 256 threads), exp-based tanh.
//  Phase 3: row softmax, keep only DIAGONAL weight (einsum 'bcc,bcd->bcd').
//  Phase 4: out = (diag * v)(16x256) @ Wout^T(256x256) via WMMA, scatter to
//           (b, c, l, d) layout with +bout.
// Vb/ba are uniform score shifts -> cancel in softmax -> dropped exactly.

typedef __attribute__((ext_vector_type(2))) float v2f;
typedef __attribute__((ext_vector_type(8))) float v8f;

#define CB 8
#define CC 16
#define CL 512
#define CD 256
#define NPOS (CB * CL)      // 4096
#define LDSTRIDE 260        // 16-row tile stride in floats (bank-conflict pad)

static __device__ __forceinline__ v8f wmma_f32(v2f a, v2f b, v8f c) {
  // D = A(16x4) * B(4x16) + C(16x16), all f32, wave32.
  return __builtin_amdgcn_wmma_f32_16x16x4_f32(
      /*neg_a=*/false, a, /*neg_b=*/false, b,
      /*c_mod=*/(short)0, c, /*reuse_a=*/false, /*reuse_b=*/false);
}

static __device__ __forceinline__ float fast_tanh(float x) {
  // tanh(x) = 1 - 2/(exp(2x)+1); exact limits at +/-inf, one v_exp_f32.
  float e = __expf(2.0f * x);
  return 1.0f - 2.0f * __builtin_amdgcn_rcpf(e + 1.0f);
}

__global__ __launch_bounds__(256, 1) void cattn_fused_kernel(
    const float* __restrict__ x,     // (B, C, L, D)
    const float* __restrict__ Wqkv,  // (3D, D) row-major
    const float* __restrict__ bqkv,  // (3D)
    const float* __restrict__ Vw,    // (1, D)
    const float* __restrict__ bh,    // (D)
    const float* __restrict__ Wout,  // (D, D) row-major
    const float* __restrict__ bout,  // (D)
    float* __restrict__ out) {       // (B, C, L, D)
  __shared__ float qs[CC * LDSTRIDE];   // q + bh
  __shared__ float ks[CC * LDSTRIDE];
  __shared__ float vs[CC * LDSTRIDE];
  __shared__ float vw[CD];
  __shared__ float sc[CC][CC + 1];
  __shared__ float dsh[CC];

  const int p   = blockIdx.x;       // position = bi*L + li
  const int bi  = p >> 9;           // / L
  const int li  = p & (CL - 1);
  const int tid = threadIdx.x;
  const int w      = tid >> 5;      // wave id 0..7
  const int lane   = tid & 31;
  const int laneLo = lane & 15;
  const int laneHi = lane >> 4;     // 0 or 1
  const int kOff   = 2 * laneHi;    // K sub-slot within the 16x16x4 fragment

  vw[tid] = Vw[tid];                // blockDim == D == 256

  // ---------------- Phase 1: QKV GEMM (WMMA f32) ----------------
  // A fragment source: row m = laneLo of this position's 16-channel tile.
  const float* arow =
      x + (((size_t)(bi * CC + laneLo)) * CL + li) * CD;

  v8f acc[6] = {};                  // wave covers N = [w*96, w*96+96)
  const int nBaseW = w * 96;
  const float* brow[6];
#pragma unroll
  for (int t = 0; t < 6; ++t)
    brow[t] = Wqkv + (size_t)(nBaseW + t * 16 + laneLo) * CD;

#pragma unroll 2
  for (int k = 0; k < CD; k += 4) {
    v2f a = *(const v2f*)(arow + k + kOff);
#pragma unroll
    for (int t = 0; t < 6; ++t) {
      v2f b = *(const v2f*)(brow[t] + k + kOff);
      acc[t] = wmma_f32(a, b, acc[t]);
    }
  }

  // Scatter D fragments into q/k/v LDS tiles (+bqkv, q also +bh).
#pragma unroll
  for (int t = 0; t < 6; ++t) {
    const int nc = nBaseW + t * 16;      // tile base col (never straddles 256/512)
    const int n  = nc + laneLo;
    const float bias = bqkv[n];
    float* dst;
    int col;
    float extra = 0.0f;
    if (nc < CD)            { dst = qs; col = n;           extra = bh[n]; }
    else if (nc < 2 * CD)   { dst = ks; col = n - CD; }
    else                    { dst = vs; col = n - 2 * CD; }
#pragma unroll
    for (int v = 0; v < 8; ++v) {
      const int row = v + 8 * laneHi;
      dst[row * LDSTRIDE + col] = acc[t][v] + bias + extra;
    }
  }
  __syncthreads();

  // ---------------- Phase 2: additive scores (one pair per thread) ---------
  const int si = tid >> 4;
  const int sj = tid & 15;
  const float* qrow = qs + si * LDSTRIDE;
  const float* krow = ks + sj * LDSTRIDE;
  float s = 0.0f;
#pragma unroll 4
  for (int d = 0; d < CD; ++d)
    s += vw[d] * fast_tanh(qrow[d] + krow[d]);
  sc[si][sj] = s;                    // Vb + ba: uniform shift, cancels in softmax
  __syncthreads();

  // ---------------- Phase 3: row softmax, keep diagonal ----------------
  if (tid < CC) {
    float mx = sc[tid][0];
#pragma unroll
    for (int j = 1; j < CC; ++j) mx = fmaxf(mx, sc[tid][j]);
    float sum = 0.0f;
#pragma unroll
    for (int j = 0; j < CC; ++j) sum += __expf(sc[tid][j] - mx);
    dsh[tid] = __expf(sc[tid][tid] - mx) / sum;
  }
  __syncthreads();

  // ---------------- Phase 4: out = (diag*v) @ Wout^T (WMMA f32) ------------
  const float dscale = dsh[laneLo];            // scales A rows (m = laneLo)
  const float* vrow  = vs + laneLo * LDSTRIDE;
  v8f oacc[2] = {};                            // wave covers N = [w*32, w*32+32)
  const float* wrow0 = Wout + (size_t)(w * 32 + laneLo) * CD;
  const float* wrow1 = wrow0 + 16 * CD;

#pragma unroll 2
  for (int k = 0; k < CD; k += 4) {
    v2f a = *(const v2f*)(vrow + k + kOff);
    a.x *= dscale;
    a.y *= dscale;
    v2f b0 = *(const v2f*)(wrow0 + k + kOff);
    v2f b1 = *(const v2f*)(wrow1 + k + kOff);
    oacc[0] = wmma_f32(a, b0, oacc[0]);
    oacc[1] = wmma_f32(a, b1, oacc[1]);
  }

  // Scatter to out[(bi*C + m)*L + li][n] with +bout.
#pragma unroll
  for (int t = 0; t < 2; ++t) {
    const int n = w * 32 + t * 16 + laneLo;
    const float bo = bout[n];
#pragma unroll
    for (int v = 0; v < 8; ++v) {
      const int m = v + 8 * laneHi;
      out[(((size_t)(bi * CC + m)) * CL + li) * CD + n] = oacc[t][v] + bo;
    }
  }
}

extern "C" void kernel_launch(void* const* d_in, const int* in_sizes, int n_in,
                              void* d_out, int out_size, void* d_ws, size_t ws_size,
                              hipStream_t stream) {
  const float* x    = (const float*)d_in[0];
  const float* Wqkv = (const float*)d_in[1];
  const float* bqkv = (const float*)d_in[2];
  const float* Vw   = (const float*)d_in[3];
  // d_in[4] = Vb, d_in[6] = ba: uniform score shifts, cancel in softmax.
  const float* bh   = (const float*)d_in[5];
  const float* Wout = (const float*)d_in[7];
  const float* bout = (const float*)d_in[8];
  float* out = (float*)d_out;

  (void)in_sizes; (void)n_in; (void)d_ws; (void)ws_size; (void)out_size;

  cattn_fused_kernel<<<dim3(NPOS), dim3(256), 0, stream>>>(
      x, Wqkv, bqkv, Vw, bh, Wout, bout, out);
}